// GINWithSkip_60928406061119
// MI455X (gfx1250) — compile-verified
//
#include <hip/hip_runtime.h>
#include <stdint.h>

// GIN-with-skip for MI455X (gfx1250, wave32).
// fp32 end-to-end with V_WMMA_F32_16X16X4_F32; weights staged into LDS once
// per block via the Tensor Data Mover (TENSOR_LOAD_TO_LDS, TENSORcnt), with
// DMA-inserted row padding giving a bank-conflict-free LDS pitch.

#define NNODES 50000
#define NEDGES 625000
#define DFEAT  128
#define ODIM   (3 * DFEAT)
#define KSTEPS (DFEAT / 4)   // 32 wmma k-steps
#define PITCH  136           // LDS row pitch in floats (128 data + 8 pad)
#define MTILES 5             // 16-row m-tiles per block (50000 = 625*5*16)

typedef __attribute__((ext_vector_type(2))) float v2f;
typedef __attribute__((ext_vector_type(8))) float v8f;
typedef __attribute__((ext_vector_type(4))) unsigned int u32x4;
typedef __attribute__((ext_vector_type(8))) int i32x8;
typedef __attribute__((ext_vector_type(4))) int i32x4;

#if defined(__has_builtin)
#if __has_builtin(__builtin_amdgcn_tensor_load_to_lds)
#define USE_TDM 1
#endif
#endif
#ifndef USE_TDM
#define USE_TDM 0
#endif

// ---------------------------------------------------------------- utilities

__global__ void fill_zero_kernel(float* __restrict__ p, int n) {
  int i = blockIdx.x * blockDim.x + threadIdx.x;
  if (i < n) p[i] = 0.0f;
}

// One 32-lane group per edge; each lane moves 4 contiguous floats.
__global__ void scatter_add_kernel(const float* __restrict__ h,
                                   const long long* __restrict__ src,
                                   const long long* __restrict__ dst,
                                   float* __restrict__ agg) {
  int t = blockIdx.x * blockDim.x + threadIdx.x;
  int e = t >> 5;
  int g = t & 31;
  if (e >= NEDGES) return;
  int s = (int)src[e];
  int d = (int)dst[e];
  const float4 v = ((const float4*)(h + (size_t)s * DFEAT))[g];
  float* a = agg + (size_t)d * DFEAT + (g << 2);
  atomicAdd(a + 0, v.x);
  atomicAdd(a + 1, v.y);
  atomicAdd(a + 2, v.z);
  atomicAdd(a + 3, v.w);
}

// --------------------------------------------------- weight staging (TDM)
//
// Stage the 128x128 fp32 weight matrix into LDS with row pitch PITCH floats.
// TDM path: wave 0 issues TENSOR_LOAD_TO_LDS with a 2D D# (ISA ch.8):
//   group0: count=1, lds_addr=0 (sW is the kernel's only LDS alloc),
//           global_addr=W, type=2.
//   group1: data_size=4B (code 2), pad_enable=1, pad_interval=128 dwords
//           (code 6), pad_amount=8 dwords (code 7) -> pitch 136 floats,
//           tensor_dim0=tensor_dim1=tile_dim0=tile_dim1=128, dim0_stride=128.
//   groups 2/3 (+ trailing group): zero (2D tensor, higher dims unused).
__device__ __forceinline__ void stage_weights(const float* __restrict__ W,
                                              float* sW) {
#if USE_TDM
  if (threadIdx.x < 32) {  // one wave issues the DMA (EXEC ignored by TDM)
    const uint64_t ga = (uint64_t)(uintptr_t)W;
    u32x4 g0;
    g0[0] = 1u;                                   // count = 1 (valid)
    g0[1] = 0u;                                   // lds_addr = 0
    g0[2] = (unsigned int)(ga & 0xffffffffu);     // global_addr[31:0]
    g0[3] = (unsigned int)((ga >> 32) & 0x01ffffffu) | 0x80000000u; // type=2
    i32x8 g1;
    g1[0] = 0x0F920000;       // data_size=2(4B) | pad_en | interval=6 | amount=7
    g1[1] = DFEAT << 16;      // tensor_dim0 = 128 (bits 79:48)
    g1[2] = DFEAT << 16;      // tensor_dim1 = 128 (bits 111:80)
    g1[3] = DFEAT << 16;      // tile_dim0   = 128 (bits 127:112)
    g1[4] = DFEAT;            // tile_dim1   = 128 (bits 143:128)
    g1[5] = DFEAT;            // tensor_dim0_stride = 128 (bits 207:160)
    g1[6] = 0;
    g1[7] = 0;
    i32x4 gz4 = {0, 0, 0, 0};
    i32x8 gz8 = {0, 0, 0, 0, 0, 0, 0, 0};
    // amdgpu-toolchain (clang-23) 6-arg form:
    // (u32x4, i32x8, i32x4, i32x4, i32x8, i32 cpol)
    __builtin_amdgcn_tensor_load_to_lds(g0, g1, gz4, gz4, gz8, 0);
    __builtin_amdgcn_s_wait_tensorcnt(0);
  }
  // The TDM builtin carries no pointer operand (the LDS address is plain
  // integer data in the descriptor), so without this escape the compiler
  // proves sW is never written and folds all LDS reads to undef (observed:
  // ds=0, all wmma B operands collapsed). Force sW to escape into an opaque
  // asm with a memory clobber so the DMA's LDS writes are assumed.
  asm volatile("" : : "v"(sW) : "memory");
  __syncthreads();
#else
  // Fallback: cooperative copy with explicit padded pitch.
  const float4* Wg = (const float4*)W;
  for (int idx = threadIdx.x; idx < DFEAT * (DFEAT / 4); idx += 256) {
    const int r = idx >> 5;   // row 0..127
    const int c = idx & 31;   // float4 col 0..31
    *(float4*)&sW[r * PITCH + c * 4] = Wg[r * 32 + c];
  }
  __syncthreads();
#endif
}

// Preload this wave's 32 B fragments (4x16 each) from LDS into registers.
// b.x = W[kk][col], b.y = W[kk+1][col]; kk = 4i + 2*half (ISA 7.12.2 layout).
// Pitch 136: lane halves hit disjoint bank sets -> conflict-free ds_load.
__device__ __forceinline__ void load_b_frags(const float* sW, int half,
                                             int col, v2f* Bf) {
#pragma unroll
  for (int i = 0; i < KSTEPS; ++i) {
    const int kk = 4 * i + 2 * half;
    v2f b;
    b.x = sW[kk * PITCH + col];
    b.y = sW[(kk + 1) * PITCH + col];
    Bf[i] = b;
  }
}

// ------------------------------------------------------------- GEMM kernels
// Block = 8 waves; wave w owns output cols [16w,16w+16); block covers MTILES
// 16-row tiles, reusing the LDS-staged W and register-resident B fragments.

// t1 = relu(((1+eps)*h + agg) @ W1 + b1)
__global__ void __launch_bounds__(256)
mlp1_wmma_kernel(const float* __restrict__ h, const float* __restrict__ agg,
                 const float* __restrict__ W, const float* __restrict__ bias,
                 const float* __restrict__ epsv, int layer,
                 float* __restrict__ out) {
  __shared__ float sW[DFEAT * PITCH];
  stage_weights(W, sW);

  const int lane = threadIdx.x & 31;
  const int wave = threadIdx.x >> 5;
  const int half = lane >> 4;
  const int lrow = lane & 15;
  const int col = wave * 16 + lrow;

  v2f Bf[KSTEPS];
  load_b_frags(sW, half, col, Bf);

  const float se = 1.0f + epsv[layer];
  const float bv = bias[col];

  for (int mt = 0; mt < MTILES; ++mt) {
    const int m0 = (blockIdx.x * MTILES + mt) * 16;
    const float* hrow = h + (size_t)(m0 + lrow) * DFEAT;
    const float* arow = agg + (size_t)(m0 + lrow) * DFEAT;
    v8f acc = {};
#pragma unroll
    for (int i = 0; i < KSTEPS; ++i) {
      const int kk = 4 * i + 2 * half;
      const float2 hv = *(const float2*)(hrow + kk);
      const float2 av = *(const float2*)(arow + kk);
      v2f a;
      a.x = se * hv.x + av.x;
      a.y = se * hv.y + av.y;
      acc = __builtin_amdgcn_wmma_f32_16x16x4_f32(false, a, false, Bf[i],
                                                  (short)0, acc, false, false);
    }
    float* optr = out + (size_t)(m0 + 8 * half) * DFEAT + col;
#pragma unroll
    for (int r = 0; r < 8; ++r)
      optr[(size_t)r * DFEAT] = fmaxf(acc[r] + bv, 0.0f);
  }
}

// z = A @ W2 + b2 ; accumulate per-feature Sum(z), Sum(z^2) for BatchNorm.
__global__ void __launch_bounds__(256)
mlp2_wmma_kernel(const float* __restrict__ A, const float* __restrict__ W,
                 const float* __restrict__ bias, float* __restrict__ z,
                 float* __restrict__ bnsum, float* __restrict__ bnsumsq) {
  __shared__ float sW[DFEAT * PITCH];
  stage_weights(W, sW);

  const int lane = threadIdx.x & 31;
  const int wave = threadIdx.x >> 5;
  const int half = lane >> 4;
  const int lrow = lane & 15;
  const int col = wave * 16 + lrow;

  v2f Bf[KSTEPS];
  load_b_frags(sW, half, col, Bf);

  const float bv = bias[col];
  float s = 0.0f, sq = 0.0f;

  for (int mt = 0; mt < MTILES; ++mt) {
    const int m0 = (blockIdx.x * MTILES + mt) * 16;
    const float* arow = A + (size_t)(m0 + lrow) * DFEAT;
    v8f acc = {};
#pragma unroll
    for (int i = 0; i < KSTEPS; ++i) {
      const int kk = 4 * i + 2 * half;
      const float2 av = *(const float2*)(arow + kk);
      v2f a;
      a.x = av.x;
      a.y = av.y;
      acc = __builtin_amdgcn_wmma_f32_16x16x4_f32(false, a, false, Bf[i],
                                                  (short)0, acc, false, false);
    }
    float* zptr = z + (size_t)(m0 + 8 * half) * DFEAT + col;
#pragma unroll
    for (int r = 0; r < 8; ++r) {
      const float v = acc[r] + bv;
      zptr[(size_t)r * DFEAT] = v;
      s += v;
      sq += v * v;
    }
  }
  atomicAdd(bnsum + col, s);
  atomicAdd(bnsumsq + col, sq);
}

// out[:, layer*128 : (layer+1)*128] = A @ Wl + bl   (row stride ODIM = 384)
__global__ void __launch_bounds__(256)
head_wmma_kernel(const float* __restrict__ A, const float* __restrict__ W,
                 const float* __restrict__ bias, float* __restrict__ out,
                 int layer) {
  __shared__ float sW[DFEAT * PITCH];
  stage_weights(W, sW);

  const int lane = threadIdx.x & 31;
  const int wave = threadIdx.x >> 5;
  const int half = lane >> 4;
  const int lrow = lane & 15;
  const int col = wave * 16 + lrow;

  v2f Bf[KSTEPS];
  load_b_frags(sW, half, col, Bf);

  const float bv = bias[col];

  for (int mt = 0; mt < MTILES; ++mt) {
    const int m0 = (blockIdx.x * MTILES + mt) * 16;
    const float* arow = A + (size_t)(m0 + lrow) * DFEAT;
    v8f acc = {};
#pragma unroll
    for (int i = 0; i < KSTEPS; ++i) {
      const int kk = 4 * i + 2 * half;
      const float2 av = *(const float2*)(arow + kk);
      v2f a;
      a.x = av.x;
      a.y = av.y;
      acc = __builtin_amdgcn_wmma_f32_16x16x4_f32(false, a, false, Bf[i],
                                                  (short)0, acc, false, false);
    }
    float* optr =
        out + (size_t)(m0 + 8 * half) * ODIM + layer * DFEAT + col;
#pragma unroll
    for (int r = 0; r < 8; ++r)
      optr[(size_t)r * ODIM] = acc[r] + bv;
  }
}

// BatchNorm (biased batch stats) + optional residual + ReLU; in-place safe.
__global__ void bn_skip_relu_kernel(const float* __restrict__ z,
                                    const float* __restrict__ bnsum,
                                    const float* __restrict__ bnsumsq,
                                    const float* __restrict__ gamma,
                                    const float* __restrict__ beta,
                                    const float* __restrict__ skip,
                                    float* __restrict__ hout, int use_skip) {
  const int i = blockIdx.x * blockDim.x + threadIdx.x;
  const int f = i & (DFEAT - 1);
  const float inv_n = 1.0f / (float)NNODES;
  const float mu = bnsum[f] * inv_n;
  const float var = bnsumsq[f] * inv_n - mu * mu;
  float v = gamma[f] * (z[i] - mu) * rsqrtf(var + 1e-5f) + beta[f];
  if (use_skip) v += skip[i];
  hout[i] = fmaxf(v, 0.0f);
}

// ------------------------------------------------------------------- launch

extern "C" void kernel_launch(void* const* d_in, const int* in_sizes, int n_in,
                              void* d_out, int out_size, void* d_ws,
                              size_t ws_size, hipStream_t stream) {
  (void)in_sizes; (void)n_in; (void)out_size; (void)ws_size;

  const float* x      = (const float*)d_in[0];
  const long long* ei = (const long long*)d_in[1];  // int64 [2, E]
  const float* W1     = (const float*)d_in[2];
  const float* b1     = (const float*)d_in[3];
  const float* W2     = (const float*)d_in[4];
  const float* b2     = (const float*)d_in[5];
  const float* eps    = (const float*)d_in[6];
  const float* gamma  = (const float*)d_in[7];
  const float* beta   = (const float*)d_in[8];
  const float* Wl     = (const float*)d_in[9];
  const float* bl     = (const float*)d_in[10];
  float* out = (float*)d_out;
  float* ws  = (float*)d_ws;

  const long long* src = ei;
  const long long* dst = ei + NEDGES;

  const size_t nd = (size_t)NNODES * DFEAT;  // 6.4M floats
  float* agg = ws;            // N*D (reused as z after GEMM1 consumes it)
  float* t1  = ws + nd;       // N*D
  float* hA  = ws + 2 * nd;   // N*D (layer activations, in-place reuse)
  float* bns = ws + 3 * nd;   // D
  float* bnq = bns + DFEAT;   // D

  const int TPB = 256;                               // 8 waves (wave32)
  const int gemm_blocks = NNODES / (16 * MTILES);    // 625 (exact)
  const int nd_blocks   = (int)(nd / TPB);           // 25000 (exact)
  const int sc_blocks   = (NEDGES * 32) / TPB;       // 78125 (exact)

  const float* hin = x;
  for (int i = 0; i < 3; ++i) {
    fill_zero_kernel<<<nd_blocks, TPB, 0, stream>>>(agg, (int)nd);
    fill_zero_kernel<<<1, TPB, 0, stream>>>(bns, 2 * DFEAT);

    scatter_add_kernel<<<sc_blocks, TPB, 0, stream>>>(hin, src, dst, agg);

    mlp1_wmma_kernel<<<gemm_blocks, TPB, 0, stream>>>(
        hin, agg, W1 + (size_t)i * DFEAT * DFEAT, b1 + i * DFEAT, eps, i, t1);

    mlp2_wmma_kernel<<<gemm_blocks, TPB, 0, stream>>>(
        t1, W2 + (size_t)i * DFEAT * DFEAT, b2 + i * DFEAT, agg /* z */, bns,
        bnq);

    bn_skip_relu_kernel<<<nd_blocks, TPB, 0, stream>>>(
        agg, bns, bnq, gamma + i * DFEAT, beta + i * DFEAT, hin, hA,
        (i > 0) ? 1 : 0);

    head_wmma_kernel<<<gemm_blocks, TPB, 0, stream>>>(
        hA, Wl + (size_t)i * DFEAT * DFEAT, bl + i * DFEAT, out, i);

    hin = hA;  // layers 1,2 run BN in-place on hA (elementwise safe)
  }
}